// BILSTM_CRF_2070174237079
// MI455X (gfx1250) — compile-verified
//
#include <hip/hip_runtime.h>
#include <hip/hip_bf16.h>
#include <stdint.h>

#define S_LEN 512
#define B_SZ  512
#define T_TAG 64

typedef __attribute__((ext_vector_type(2))) float v2f;
typedef __attribute__((ext_vector_type(8))) float v8f;

#ifndef __has_builtin
#define __has_builtin(x) 0
#endif

// ---------------- CDNA5 async-to-LDS helpers (probe-verified) ----------------
// Probe result (round 1): builtins exist; param0 is addrspace(1) TYPED ptr
// (int* for b32, char* for b8), param1 is the LDS ptr, then imm offset, imm cpol.
#if defined(__gfx1250__) && __has_builtin(__builtin_amdgcn_global_load_async_to_lds_b32)
#define HAVE_ASYNC_B32 1
#else
#define HAVE_ASYNC_B32 0
#endif

#if defined(__gfx1250__) && __has_builtin(__builtin_amdgcn_global_load_async_to_lds_b8)
#define HAVE_ASYNC_B8 1
#else
#define HAVE_ASYNC_B8 0
#endif

#if defined(__gfx1250__)
# if __has_builtin(__builtin_amdgcn_s_wait_asynccnt)
#  define WAIT_ASYNC(n) __builtin_amdgcn_s_wait_asynccnt(n)
# else
#  define WAIT_ASYNC(n) asm volatile("s_wait_asynccnt %0" ::"i"(n))
# endif
#else
# define WAIT_ASYNC(n) ((void)0)
#endif

__device__ __forceinline__ void async_copy_b32(const void* gsrc, void* ldst) {
#if HAVE_ASYNC_B32
  __builtin_amdgcn_global_load_async_to_lds_b32(
      (__attribute__((address_space(1))) int*)(void*)gsrc,
      (__attribute__((address_space(3))) int*)ldst, 0, 0);
#else
  *(float*)ldst = *(const float*)gsrc;  // sync fallback (correctness-equivalent)
#endif
}

__device__ __forceinline__ void async_copy_b8(const void* gsrc, void* ldst) {
#if HAVE_ASYNC_B8
  __builtin_amdgcn_global_load_async_to_lds_b8(
      (__attribute__((address_space(1))) char*)(void*)gsrc,
      (__attribute__((address_space(3))) char*)ldst, 0, 0);
#else
  *(uint8_t*)ldst = *(const uint8_t*)gsrc;  // sync fallback
#endif
}

// ---------------------------------------------------------------------------
// Kernel 0: lengths[b] = sum_s mask[b,s]  via  V_WMMA_F32_16X16X4_F32
// (the one true matmul in this workload: mask(512x512) @ ones(512))
// A 16x4 f32 layout: lanes 0-15 M=0..15 {v0:K=0, v1:K=1};
//                    lanes 16-31 M=0..15 {v0:K=2, v1:K=3}.
// C 16x16 f32 layout: vgpr r, lanes 0-15 -> (M=r, N=lane);
//                     lanes 16-31 -> (M=r+8, N=lane-16).
// ---------------------------------------------------------------------------
__global__ __launch_bounds__(256) void lengths_wmma_kernel(
    const float* __restrict__ mask, int* __restrict__ lengths) {
  const int wave = threadIdx.x >> 5;
  const int lane = threadIdx.x & 31;
  const int m0 = (blockIdx.x * 8 + wave) * 16;  // 4 blocks * 8 waves * 16 rows = 512
  const int row = m0 + (lane & 15);
  const int koff = (lane >> 4) * 2;

  v8f acc = {};
  v2f bones;
  bones.x = 1.0f;
  bones.y = 1.0f;

  const float* rp = mask + (size_t)row * S_LEN + koff;
#pragma unroll 4
  for (int kb = 0; kb < S_LEN; kb += 4) {
    v2f a;
    a.x = rp[kb + 0];
    a.y = rp[kb + 1];
    acc = __builtin_amdgcn_wmma_f32_16x16x4_f32(
        /*neg_a=*/false, a, /*neg_b=*/false, bones,
        /*c_mod=*/(short)0, acc, /*reuse_a=*/false, /*reuse_b=*/false);
  }
  // every column of C holds the row sums; pull column 0 from lanes 0 / 16
  if (lane == 0) {
#pragma unroll
    for (int r = 0; r < 8; ++r) lengths[m0 + r] = (int)(acc[r] + 0.5f);
  }
  if (lane == 16) {
#pragma unroll
    for (int r = 0; r < 8; ++r) lengths[m0 + 8 + r] = (int)(acc[r] + 0.5f);
  }
}

// ---------------------------------------------------------------------------
// Kernel 1: Viterbi forward scan. 4 batches per 256-thread block (64 thr/batch,
// thread j owns tag j). transitions (16KB) in LDS (bank-conflict free:
// s_trans[i*64+j] -> bank j). v double-buffered in LDS; per-step emission rows
// streamed ahead with async-to-LDS double buffering. Backpointers packed u8.
// ---------------------------------------------------------------------------
__global__ __launch_bounds__(256) void viterbi_fwd_kernel(
    const float* __restrict__ feats, const float* __restrict__ mask,
    const float* __restrict__ startT, const float* __restrict__ endT,
    const float* __restrict__ trans, const int* __restrict__ lengths,
    uint8_t* __restrict__ hist, int* __restrict__ last_path) {
  __shared__ float s_trans[T_TAG * T_TAG];
  __shared__ float s_v[2][4][T_TAG];
  __shared__ float s_emis[2][4][T_TAG];
  __shared__ float s_fin[4][T_TAG];
  __shared__ int s_lp[4];
  __shared__ int s_L[4];

  const int tid = threadIdx.x;
  const int g = tid >> 6;   // sub-batch within block
  const int j = tid & 63;   // tag
  const int b = blockIdx.x * 4 + g;

  for (int k = tid; k < T_TAG * T_TAG; k += 256) s_trans[k] = trans[k];

  // s = 0: score0 = start_transitions + feats[0]
  float vj = startT[j] + feats[(size_t)b * T_TAG + j];
  s_v[0][g][j] = vj;

  // prefetch emissions for s = 1 into buffer (1 & 1)
  async_copy_b32(feats + ((size_t)1 * B_SZ + b) * T_TAG + j, &s_emis[1][g][j]);
  __syncthreads();

  for (int s = 1; s < S_LEN; ++s) {
    const int rb = (s - 1) & 1;  // v read buffer
    const int wb = s & 1;        // v write buffer == emis buffer for step s

    // prefetch emissions for s+1 (other emis buffer)
    if (s + 1 < S_LEN) {
      async_copy_b32(feats + ((size_t)(s + 1) * B_SZ + b) * T_TAG + j,
                     &s_emis[(s + 1) & 1][g][j]);
    }

    // max-plus GEMV column j: best over i of v[i] + trans[i][j] (first-max tie)
    float best = -__builtin_inff();
    int arg = 0;
#pragma unroll 8
    for (int i = 0; i < T_TAG; ++i) {
      float t = s_v[rb][g][i] + s_trans[i * T_TAG + j];
      if (t > best) {
        best = t;
        arg = i;
      }
    }

    // retire the async load that carries this step's emissions
    if (s + 1 < S_LEN) {
      WAIT_ASYNC(1);
    } else {
      WAIT_ASYNC(0);
    }

    const float em = s_emis[wb][g][j];
    const float m = mask[(size_t)b * S_LEN + s];
    const float bt = best + em;
    const float nv = bt * m + vj * (1.0f - m);
    hist[((size_t)(s - 1) * B_SZ + b) * T_TAG + j] =
        (uint8_t)((m != 0.0f) ? arg : 0);
    s_v[wb][g][j] = nv;
    vj = nv;
    __syncthreads();
  }

  // final_v = v + end_transitions ; last_path = first-argmax over tags
  s_fin[g][j] = vj + endT[j];
  __syncthreads();
  if (j == 0) {
    float bv = s_fin[g][0];
    int bi = 0;
    for (int k = 1; k < T_TAG; ++k) {
      float f = s_fin[g][k];
      if (f > bv) {
        bv = f;
        bi = k;
      }
    }
    s_lp[g] = bi;
    s_L[g] = lengths[b];
    last_path[b] = bi;
  }
  __syncthreads();
  const int L = s_L[g];
  const int lp = s_lp[g];
  // hb[b, L-1, :] = last_path  (row S-1 is never read by the backtrace)
  hist[((size_t)(L - 1) * B_SZ + b) * T_TAG + j] = (uint8_t)lp;
}

// ---------------------------------------------------------------------------
// Kernel 2: backtrace. One 64-thread block per batch. Row addresses are
// independent of the pointer chase, so prefetch whole 64B history rows with a
// depth-2 async-to-LDS pipeline; the dependent op is one LDS byte read.
// ---------------------------------------------------------------------------
__global__ __launch_bounds__(64) void backtrace_kernel(
    const uint8_t* __restrict__ hist, const int* __restrict__ last_path,
    int* __restrict__ out) {
  __shared__ uint8_t rows[2][T_TAG];
  const int b = blockIdx.x;
  const int j = threadIdx.x;

  // preload rows S-2 (buf 0) and S-3 (buf 1)
  async_copy_b8(&hist[((size_t)(S_LEN - 2) * B_SZ + b) * T_TAG + j],
                &rows[(S_LEN - 2) & 1][j]);
  async_copy_b8(&hist[((size_t)(S_LEN - 3) * B_SZ + b) * T_TAG + j],
                &rows[(S_LEN - 3) & 1][j]);

  int lp = 0;
  if (j == 0) {
    lp = last_path[b];
    out[(size_t)b * S_LEN + (S_LEN - 1)] = lp;
  }

  for (int s = S_LEN - 2; s >= 0; --s) {
    if (s >= 1) {
      WAIT_ASYNC(1);  // oldest outstanding row (row s) has landed
    } else {
      WAIT_ASYNC(0);
    }
    __syncthreads();  // make both waves' async bytes visible to thread 0
    if (j == 0) {
      lp = rows[s & 1][lp];
      out[(size_t)b * S_LEN + s] = lp;
    }
    __syncthreads();  // protect buffer reuse (WAR) before re-issuing
    if (s >= 2) {
      async_copy_b8(&hist[((size_t)(s - 2) * B_SZ + b) * T_TAG + j],
                    &rows[s & 1][j]);
    }
  }
}

// ---------------------------------------------------------------------------
extern "C" void kernel_launch(void* const* d_in, const int* in_sizes, int n_in,
                              void* d_out, int out_size, void* d_ws,
                              size_t ws_size, hipStream_t stream) {
  const float* feats = (const float*)d_in[0];   // (S,B,T) f32
  const float* mask = (const float*)d_in[1];    // (B,S)  f32 0/1
  const float* startT = (const float*)d_in[2];  // (T,)
  const float* endT = (const float*)d_in[3];    // (T,)
  const float* trans = (const float*)d_in[4];   // (T,T)
  int* out = (int*)d_out;                       // (B,S) int32

  // workspace layout
  const size_t histBytes = (size_t)S_LEN * B_SZ * T_TAG;  // 16 MB, u8 bp
  uint8_t* hist = (uint8_t*)d_ws;
  int* last_path = (int*)((char*)d_ws + histBytes);  // [B]
  int* lengths = last_path + B_SZ;                   // [B]

  lengths_wmma_kernel<<<4, 256, 0, stream>>>(mask, lengths);
  viterbi_fwd_kernel<<<B_SZ / 4, 256, 0, stream>>>(
      feats, mask, startT, endT, trans, lengths, hist, last_path);
  backtrace_kernel<<<B_SZ, 64, 0, stream>>>(hist, last_path, out);
}